// GCNConvManual_67095979098876
// MI455X (gfx1250) — compile-verified
//
#include <hip/hip_runtime.h>
#include <hip/hip_bf16.h>

// ---------------------------------------------------------------------------
// GCNConv on MI455X (gfx1250):
//   xt  = x @ W                (fp32 WMMA 16x16x4, exact fp32 accumulate)
//   deg = scatter_add(1, dst) + 1 (self loop);  dis = rsqrt(deg)
//   out = bias + xt*dis^2 (self loop)  +  scatter_add(xt[src]*dis[s]*dis[d], dst)
// Memory-bound on the edge gather/scatter (~1 GB traffic); GEMM is 1.64 GFLOP.
// ---------------------------------------------------------------------------

typedef float v2f __attribute__((ext_vector_type(2)));
typedef float v8f __attribute__((ext_vector_type(8)));

#define IN_C 128
#define OUT_C 128

// ---- 1. deg[i] = 1.0 (self-loop) ------------------------------------------
__global__ void gcn_init_deg(float* __restrict__ deg, int n) {
    int i = blockIdx.x * blockDim.x + threadIdx.x;
    if (i < n) deg[i] = 1.0f;
}

// ---- 2. deg[dst[e]] += 1 ---------------------------------------------------
__global__ void gcn_accum_deg(const int* __restrict__ dst,
                              float* __restrict__ deg, int e_cnt) {
    int e = blockIdx.x * blockDim.x + threadIdx.x;
    if (e < e_cnt) atomicAdd(&deg[dst[e]], 1.0f);
}

// ---- 3. dis = rsqrt(deg) in place -----------------------------------------
__global__ void gcn_rsqrt_deg(float* __restrict__ deg, int n) {
    int i = blockIdx.x * blockDim.x + threadIdx.x;
    if (i < n) {
        float d = deg[i];
        deg[i] = (d > 0.0f) ? rsqrtf(d) : 0.0f;
    }
}

// ---- 4. xt = x @ W via V_WMMA_F32_16X16X4_F32 ------------------------------
// One wave computes a 16x16 tile of xt. Block = 256 threads = 8 waves = 8 col
// tiles (covers all 128 cols of one 16-row strip). N = 50000 = 3125 * 16, so
// every wave is full (EXEC all-ones, as WMMA requires).
//
// Fragment layouts (ISA 7.12.2, 32-bit operands, wave32):
//   A 16x4:  lanes 0-15 -> M=lane, K={0,1} in VGPR{0,1};
//            lanes 16-31 -> M=lane-16, K={2,3}          (contiguous float2)
//   B 4x16:  lanes 0-15 -> N=lane, K={0,1}; lanes 16-31 -> N=lane-16, K={2,3}
//   C/D 16x16: VGPR v -> row v (lanes 0-15) / row v+8 (lanes 16-31), col lane&15
__global__ __launch_bounds__(256) void gcn_gemm_wmma(
    const float* __restrict__ X, const float* __restrict__ W,
    float* __restrict__ XT) {
    const int lane = threadIdx.x & 31;
    const int wave = threadIdx.x >> 5;
    const int m0   = blockIdx.x * 16;
    const int n0   = wave * 16;
    const int l15  = lane & 15;
    const int hi   = lane >> 4;      // 0: lanes 0-15, 1: lanes 16-31
    const int koff = hi * 2;

    const float* __restrict__ arow = X + (size_t)(m0 + l15) * IN_C;

    v8f acc = {};
#pragma unroll 4
    for (int k0 = 0; k0 < IN_C; k0 += 4) {
        v2f a = *(const v2f*)(arow + k0 + koff);             // A[M=l15][k0+koff .. +1]
        v2f b;
        b.x = W[(size_t)(k0 + koff + 0) * OUT_C + n0 + l15]; // B[k][N=l15]
        b.y = W[(size_t)(k0 + koff + 1) * OUT_C + n0 + l15];
        // (neg_a, A, neg_b, B, c_mod, C, reuse_a, reuse_b)
        acc = __builtin_amdgcn_wmma_f32_16x16x4_f32(
            false, a, false, b, (short)0, acc, false, false);
    }

    float* __restrict__ o = XT + (size_t)(m0 + hi * 8) * OUT_C + n0 + l15;
#pragma unroll
    for (int v = 0; v < 8; ++v)
        o[(size_t)v * OUT_C] = acc[v];
}

// ---- 5. out = bias + xt * dis^2 (self-loop message) ------------------------
__global__ void gcn_init_out(const float* __restrict__ xt,
                             const float* __restrict__ dis,
                             const float* __restrict__ bias,
                             float* __restrict__ out, long long total) {
    long long idx = (long long)blockIdx.x * blockDim.x + threadIdx.x;
    if (idx >= total) return;
    int i = (int)(idx >> 7);      // node
    int c = (int)(idx & 127);     // channel
    float di = dis[i];
    out[idx] = bias[c] + xt[idx] * di * di;
}

// ---- 6. out[dst] += xt[src] * dis[src]*dis[dst] ----------------------------
// 32 threads per edge; each thread handles 4 channels (float4 gather,
// 4x global_atomic_add_f32 scatter).
__global__ void gcn_scatter(const int* __restrict__ src,
                            const int* __restrict__ dst,
                            const float* __restrict__ dis,
                            const float* __restrict__ xt,
                            float* __restrict__ out, int e_cnt) {
    long long idx = (long long)blockIdx.x * blockDim.x + threadIdx.x;
    int e  = (int)(idx >> 5);
    int c4 = (int)(idx & 31);
    if (e >= e_cnt) return;
    int s = src[e];
    int d = dst[e];
    float norm = dis[s] * dis[d];
    const float4 m = *(const float4*)(xt + (size_t)s * OUT_C + c4 * 4);
    float* o = out + (size_t)d * OUT_C + c4 * 4;
    atomicAdd(o + 0, m.x * norm);
    atomicAdd(o + 1, m.y * norm);
    atomicAdd(o + 2, m.z * norm);
    atomicAdd(o + 3, m.w * norm);
}

extern "C" void kernel_launch(void* const* d_in, const int* in_sizes, int n_in,
                              void* d_out, int out_size, void* d_ws, size_t ws_size,
                              hipStream_t stream) {
    const float* x    = (const float*)d_in[0];   // [N, 128]
    const int*   ei   = (const int*)d_in[1];     // [2, E]
    const float* W    = (const float*)d_in[2];   // [128, 128]
    const float* bias = (const float*)d_in[3];   // [128]
    float*       out  = (float*)d_out;           // [N, 128]

    const int N = in_sizes[0] / IN_C;            // 50000
    const int E = in_sizes[1] / 2;               // 800000
    const int* src = ei;
    const int* dst = ei + E;

    // workspace: deg/dis (N floats, padded to 256) | xt (N*128 floats)
    float* deg = (float*)d_ws;
    float* xt  = deg + (((size_t)N + 255) & ~(size_t)255);

    const int T = 256;

    gcn_init_deg<<<(N + T - 1) / T, T, 0, stream>>>(deg, N);
    gcn_accum_deg<<<(E + T - 1) / T, T, 0, stream>>>(dst, deg, E);
    gcn_rsqrt_deg<<<(N + T - 1) / T, T, 0, stream>>>(deg, N);

    // N = 3125 * 16 exactly: one block per 16-row strip, 8 waves = 8 col tiles
    gcn_gemm_wmma<<<N / 16, 256, 0, stream>>>(x, W, xt);

    long long total = (long long)N * OUT_C;
    gcn_init_out<<<(unsigned)((total + T - 1) / T), T, 0, stream>>>(xt, deg, bias, out, total);

    long long sthreads = (long long)E * 32;
    gcn_scatter<<<(unsigned)((sthreads + T - 1) / T), T, 0, stream>>>(src, dst, deg, xt, out, E);
}